// GAT_1090921693566
// MI455X (gfx1250) — compile-verified
//
#include <hip/hip_runtime.h>

#define N_  50000
#define E_  800000
#define D_  256
#define H_  4
#define F_  32
#define HF_ 128
#define NEG_SLOPE 0.2f

typedef __attribute__((ext_vector_type(2))) float v2f;
typedef __attribute__((ext_vector_type(8))) float v8f;

// ---- order-preserving float<->uint encoding for atomic max on signed floats ----
__device__ __forceinline__ unsigned encodeOrd(float f) {
  unsigned u = __float_as_uint(f);
  return (u & 0x80000000u) ? ~u : (u | 0x80000000u);
}
__device__ __forceinline__ float decodeOrd(unsigned u) {
  unsigned b = (u & 0x80000000u) ? (u & 0x7FFFFFFFu) : ~u;
  return __uint_as_float(b);
}

// ============================================================================
// Kernel 1: hW = h @ W  via V_WMMA_F32_16X16X4_F32
// Each wave computes a 16-row x 128-col strip. 8 waves per block (wave32).
// A layout (16x4 f32): lane L -> M = L%16, VGPR0/1 = K = 2*(L/16) + {0,1}
// B layout (4x16 f32): lane L -> N = L%16, VGPR0/1 = K = 2*(L/16) + {0,1}
// D layout (16x16 f32): VGPR v, lane L -> M = v + 8*(L/16), N = L%16
// ============================================================================
__global__ __launch_bounds__(256) void gat_gemm_wmma(
    const float* __restrict__ h, const float* __restrict__ W,
    float* __restrict__ hW, int nRows)
{
  const int wave = threadIdx.x >> 5;
  const int lane = threadIdx.x & 31;
  const int rowBase = (blockIdx.x * 8 + wave) * 16;
  if (rowBase >= nRows) return;          // wave-uniform: EXEC stays all-ones

  const int m    = lane & 15;
  const int khalf= (lane >> 4) * 2;      // 0 or 2

  v8f acc[8];
#pragma unroll
  for (int t = 0; t < 8; ++t) acc[t] = (v8f){0.f,0.f,0.f,0.f,0.f,0.f,0.f,0.f};

  for (int k0 = 0; k0 < D_; k0 += 4) {
    // A fragment: 16x4 strip of h
    const float* ap = h + (size_t)(rowBase + m) * D_ + (k0 + khalf);
    v2f a; a.x = ap[0]; a.y = ap[1];
#pragma unroll
    for (int t = 0; t < 8; ++t) {
      const int n = t * 16 + (lane & 15);
      const float* bp = W + (size_t)(k0 + khalf) * HF_ + n;
      v2f b; b.x = bp[0]; b.y = bp[HF_];
      acc[t] = __builtin_amdgcn_wmma_f32_16x16x4_f32(
          /*neg_a=*/false, a, /*neg_b=*/false, b,
          /*c_mod=*/(short)0, acc[t], /*reuse_a=*/false, /*reuse_b=*/false);
    }
  }

  const int mHalf = (lane >> 4) * 8;
#pragma unroll
  for (int t = 0; t < 8; ++t) {
    const int n = t * 16 + (lane & 15);
#pragma unroll
    for (int v = 0; v < 8; ++v) {
      hW[(size_t)(rowBase + mHalf + v) * HF_ + n] = acc[t][v];
    }
  }
}

// ============================================================================
// Kernel 2: el/er per (node, head): dot(hW[n,h,:], attn_{l,r}_w) + bias
// ============================================================================
__global__ __launch_bounds__(256) void gat_attn_proj(
    const float* __restrict__ hW,
    const float* __restrict__ alw, const float* __restrict__ alb,
    const float* __restrict__ arw, const float* __restrict__ arb,
    float* __restrict__ el, float* __restrict__ er, int nNH)
{
  const int idx = blockIdx.x * blockDim.x + threadIdx.x;   // (node*H + head)
  if (idx >= nNH) return;
  const float* p = hW + (size_t)idx * F_;
  float sl = 0.f, sr = 0.f;
#pragma unroll
  for (int f = 0; f < F_; ++f) {
    const float x = p[f];
    sl += x * alw[f];
    sr += x * arw[f];
  }
  el[idx] = sl + alb[0];
  er[idx] = sr + arb[0];
}

// ============================================================================
// Kernel 3: init emax (encoded -inf), denom (0), out (bias broadcast)
// ============================================================================
__global__ __launch_bounds__(256) void gat_init(
    unsigned* __restrict__ emax, float* __restrict__ denom,
    float* __restrict__ out, const float* __restrict__ bias,
    int nNH, int nNHF)
{
  const int i = blockIdx.x * blockDim.x + threadIdx.x;
  if (i < nNH) {
    emax[i]  = encodeOrd(-__builtin_inff());
    denom[i] = 0.f;
  }
  if (i < nNHF) out[i] = bias[i & (HF_ - 1)];
}

// ============================================================================
// Kernel 4: edge logits -> leaky relu -> segment max over dst (atomic on enc)
// one thread per (edge, head)
// ============================================================================
__global__ __launch_bounds__(256) void gat_edge_max(
    const int* __restrict__ src, const int* __restrict__ dst,
    const float* __restrict__ el, const float* __restrict__ er,
    unsigned* __restrict__ emax, int nEH)
{
  const int i = blockIdx.x * blockDim.x + threadIdx.x;
  if (i >= nEH) return;
  const int e = i >> 2;          // / H_
  const int hh = i & (H_ - 1);
  const int s = src[e], d = dst[e];
  float v = er[s * H_ + hh] + el[d * H_ + hh];
  v = (v >= 0.f) ? v : NEG_SLOPE * v;
  atomicMax(&emax[d * H_ + hh], encodeOrd(v));
}

// ============================================================================
// Kernel 5: ee = exp(e - emax[dst]); denom[dst] += ee
// ============================================================================
__global__ __launch_bounds__(256) void gat_edge_exp(
    const int* __restrict__ src, const int* __restrict__ dst,
    const float* __restrict__ el, const float* __restrict__ er,
    const unsigned* __restrict__ emax, float* __restrict__ ee,
    float* __restrict__ denom, int nEH)
{
  const int i = blockIdx.x * blockDim.x + threadIdx.x;
  if (i >= nEH) return;
  const int e = i >> 2;
  const int hh = i & (H_ - 1);
  const int s = src[e], d = dst[e];
  float v = er[s * H_ + hh] + el[d * H_ + hh];
  v = (v >= 0.f) ? v : NEG_SLOPE * v;
  const float x = expf(v - decodeOrd(emax[d * H_ + hh]));
  ee[i] = x;
  atomicAdd(&denom[d * H_ + hh], x);
}

// ============================================================================
// Kernel 6: out[dst,h,f] += hW[src,h,f] * (ee[e,h] / denom[dst,h])
// one thread per (edge, h*F+f): 128 threads per edge, 2 edges per block
// ============================================================================
__global__ __launch_bounds__(256) void gat_scatter(
    const int* __restrict__ src, const int* __restrict__ dst,
    const float* __restrict__ hW, const float* __restrict__ ee,
    const float* __restrict__ denom, float* __restrict__ out,
    long long nTot)
{
  const long long tid = (long long)blockIdx.x * blockDim.x + threadIdx.x;
  if (tid >= nTot) return;
  const long long e = tid >> 7;        // / 128
  const int hf = (int)(tid & 127);
  const int hh = hf >> 5;
  const int s = src[e], d = dst[e];
  const float a = ee[e * H_ + hh] / denom[d * H_ + hh];
  atomicAdd(&out[(size_t)d * HF_ + hf], hW[(size_t)s * HF_ + hf] * a);
}

extern "C" void kernel_launch(void* const* d_in, const int* in_sizes, int n_in,
                              void* d_out, int out_size, void* d_ws, size_t ws_size,
                              hipStream_t stream)
{
  const float* h    = (const float*)d_in[0];
  const float* W    = (const float*)d_in[1];
  const float* alw  = (const float*)d_in[2];
  const float* alb  = (const float*)d_in[3];
  const float* arw  = (const float*)d_in[4];
  const float* arb  = (const float*)d_in[5];
  const float* bias = (const float*)d_in[6];
  const int*   src  = (const int*)d_in[7];
  const int*   dst  = (const int*)d_in[8];
  float* out = (float*)d_out;

  // workspace carve (256B aligned)
  char* ws = (char*)d_ws;
  size_t off = 0;
  auto carve = [&](size_t bytes) -> void* {
    void* p = ws + off;
    off += (bytes + 255) & ~(size_t)255;
    return p;
  };
  float*    hW    = (float*)carve((size_t)N_ * HF_ * sizeof(float));  // 25.6 MB
  float*    el    = (float*)carve((size_t)N_ * H_ * sizeof(float));
  float*    er    = (float*)carve((size_t)N_ * H_ * sizeof(float));
  unsigned* emax  = (unsigned*)carve((size_t)N_ * H_ * sizeof(unsigned));
  float*    denom = (float*)carve((size_t)N_ * H_ * sizeof(float));
  float*    ee    = (float*)carve((size_t)E_ * H_ * sizeof(float));   // 12.8 MB
  (void)ws_size; (void)in_sizes; (void)n_in; (void)out_size;

  // 1) GEMM: 3125 strips of 16 rows, 8 waves/block
  {
    const int strips = N_ / 16;                 // 3125
    const int blocks = (strips + 7) / 8;        // 391
    gat_gemm_wmma<<<blocks, 256, 0, stream>>>(h, W, hW, N_);
  }
  // 2) attention projections
  {
    const int nNH = N_ * H_;
    gat_attn_proj<<<(nNH + 255) / 256, 256, 0, stream>>>(hW, alw, alb, arw, arb, el, er, nNH);
  }
  // 3) init emax/denom/out
  {
    const int nNH = N_ * H_;
    const int nNHF = N_ * H_ * F_;
    gat_init<<<(nNHF + 255) / 256, 256, 0, stream>>>(emax, denom, out, bias, nNH, nNHF);
  }
  // 4) segment max
  {
    const int nEH = E_ * H_;
    gat_edge_max<<<(nEH + 255) / 256, 256, 0, stream>>>(src, dst, el, er, emax, nEH);
  }
  // 5) exp + segment sum
  {
    const int nEH = E_ * H_;
    gat_edge_exp<<<(nEH + 255) / 256, 256, 0, stream>>>(src, dst, el, er, emax, ee, denom, nEH);
  }
  // 6) weighted scatter-sum
  {
    const long long nTot = (long long)E_ * HF_;          // 102.4M
    const long long blocks = (nTot + 255) / 256;
    gat_scatter<<<(unsigned)blocks, 256, 0, stream>>>(src, dst, hW, ee, denom, out, nTot);
  }
}